// PoseNet_14388140442054
// MI455X (gfx1250) — compile-verified
//
#include <hip/hip_runtime.h>
#include <hip/hip_bf16.h>

typedef __attribute__((ext_vector_type(2))) float v2f;
typedef __attribute__((ext_vector_type(4))) float v4f;
typedef __attribute__((ext_vector_type(8))) float v8f;

#define JNT 21
#define CENC 256
#define CIMG 512
#define DBINS 64
#define NB 16

// ---------------------------------------------------------------------------
// Kernel A1: feat_x[b][c][w] = mean_h enc, feat_y[b][c][h] = mean_w enc
// one wave per (b,c); lanes 0-15 -> feat_x, lanes 16-31 -> feat_y
// ---------------------------------------------------------------------------
__global__ void k_enc_means(const float* __restrict__ enc,
                            float* __restrict__ fx, float* __restrict__ fy) {
    int bc = blockIdx.x;                 // b*256 + c
    int t  = threadIdx.x;                // 0..31
    const float* base = enc + (size_t)bc * 256;   // 16x16 tile
    if (t < 16) {
        float s = 0.f;
        #pragma unroll
        for (int h = 0; h < 16; ++h) s += base[h * 16 + t];
        fx[bc * 16 + t] = s * (1.f / 16.f);
    } else {
        int h = t - 16;
        float s = 0.f;
        #pragma unroll
        for (int w = 0; w < 16; ++w) s += base[h * 16 + w];
        fy[bc * 16 + h] = s * (1.f / 16.f);
    }
}

// ---------------------------------------------------------------------------
// Kernel A2: fz[b][o] = mean over 256 spatial elems of img_feat (o < 512)
// ---------------------------------------------------------------------------
__global__ void k_fz(const float* __restrict__ img, float* __restrict__ fz) {
    int i = blockIdx.x * blockDim.x + threadIdx.x;   // 0..8191
    if (i >= NB * CIMG) return;
    const float* base = img + (size_t)i * 256;
    float s = 0.f;
    for (int p = 0; p < 256; ++p) s += base[p];
    fz[i] = s * (1.f / 256.f);
}

// ---------------------------------------------------------------------------
// Kernel B: z = wz1[16384x512] x fz^T[512x16], + bias, BN(eval), ReLU
//           -> zbuf[batch][16384]   (o = c*64 + d)
// WMMA f32 16x16x4: one wave per 16(M=o) x 16(N=batch) tile, K=512.
// A layout (16x4 f32): lane l holds row M=l&15; VGPR0=K+2*(l>>4), VGPR1=K+1+2*(l>>4)
// B layout mirrors with N=l&15. C/D: lane l, vgpr r -> M = r + 8*(l>>4), N = l&15.
// ---------------------------------------------------------------------------
__global__ void k_zgemm(const float* __restrict__ wz1, const float* __restrict__ fz,
                        const float* __restrict__ bz1, const float* __restrict__ gamma,
                        const float* __restrict__ beta, const float* __restrict__ rmean,
                        const float* __restrict__ rvar, float* __restrict__ zbuf) {
    int lane = threadIdx.x & 31;
    int wave = threadIdx.x >> 5;
    int m    = blockIdx.x * 8 + wave;        // 0..1023 output row tiles
    int row  = m * 16 + (lane & 15);         // o row for A operand
    int col  = lane & 15;                    // batch column for B operand
    int koff = (lane >> 4) * 2;

    const float* arow = wz1 + (size_t)row * CIMG + koff;
    const float* brow = fz  + (size_t)col * CIMG + koff;

    v8f acc = {};
    #pragma unroll 4
    for (int k = 0; k < CIMG; k += 4) {
        v2f a; a.x = arow[k]; a.y = arow[k + 1];
        v2f b; b.x = brow[k]; b.y = brow[k + 1];
        acc = __builtin_amdgcn_wmma_f32_16x16x4_f32(
            /*neg_a=*/false, a, /*neg_b=*/false, b,
            /*c_mod=*/(short)0, acc, /*reuse_a=*/false, /*reuse_b=*/false);
    }

    int half  = (lane >> 4) * 8;
    int batch = lane & 15;
    #pragma unroll
    for (int r = 0; r < 8; ++r) {
        int o = m * 16 + half + r;
        float v = acc[r] + bz1[o];
        v = gamma[o] * (v - rmean[o]) * rsqrtf(rvar[o] + 1e-5f) + beta[o];
        v = v > 0.f ? v : 0.f;
        zbuf[(size_t)batch * (CENC * DBINS) + o] = v;
    }
}

// ---------------------------------------------------------------------------
// Kernel C: per (b,j) -> hx (16), hy (16), hz (64) dots + soft-argmax -> coord
// ---------------------------------------------------------------------------
__global__ void k_coords(const float* __restrict__ fx, const float* __restrict__ fy,
                         const float* __restrict__ zbuf,
                         const float* __restrict__ wx, const float* __restrict__ bx,
                         const float* __restrict__ wy, const float* __restrict__ by,
                         const float* __restrict__ wz2, const float* __restrict__ bz2,
                         float* __restrict__ out) {
    int b = blockIdx.x / JNT;
    int j = blockIdx.x % JNT;
    __shared__ float h[96];
    int t = threadIdx.x;                 // 128 threads
    if (t < 96) {
        float s;
        if (t < 16) {
            s = bx[j];
            const float* w = wx + j * CENC;
            const float* f = fx + (size_t)b * CENC * 16 + t;
            for (int c = 0; c < CENC; ++c) s += w[c] * f[c * 16];
        } else if (t < 32) {
            int hh = t - 16;
            s = by[j];
            const float* w = wy + j * CENC;
            const float* f = fy + (size_t)b * CENC * 16 + hh;
            for (int c = 0; c < CENC; ++c) s += w[c] * f[c * 16];
        } else {
            int d = t - 32;
            s = bz2[j];
            const float* w = wz2 + j * CENC;
            const float* f = zbuf + (size_t)b * (CENC * DBINS) + d;
            for (int c = 0; c < CENC; ++c) s += w[c] * f[c * DBINS];
        }
        h[t] = s;
    }
    __syncthreads();
    if (t < 3) {
        int off = (t == 0) ? 0 : (t == 1) ? 16 : 32;
        int len = (t < 2) ? 16 : 64;
        float mx = -1e30f;
        for (int i = 0; i < len; ++i) mx = fmaxf(mx, h[off + i]);
        float sum = 0.f, pos = 0.f;
        for (int i = 0; i < len; ++i) {
            float e = __expf(h[off + i] - mx);
            sum += e;
            pos += e * (float)i;
        }
        out[((size_t)b * JNT + j) * 3 + t] = pos / sum;
    }
}

// ---------------------------------------------------------------------------
// Kernel D: img_feat_xyz[b][c][h][w][d] = enc[b][c][h][w] * zbuf[b][c*64+d]
// Thread = (hw-chunk, bc, d-quad). Z quad kept in registers, NT b128 stores.
// Output: 268 MB streamed -> the run-time floor of the whole problem.
// ---------------------------------------------------------------------------
__global__ void k_bcast(const float* __restrict__ enc, const float* __restrict__ zbuf,
                        float* __restrict__ out) {
    int gid   = blockIdx.x * blockDim.x + threadIdx.x;   // 0..262143
    int chunk = gid >> 16;                               // 0..3 (hw quarter)
    int idx   = gid & 65535;
    int bc    = idx >> 4;                                // 0..4095 (b*256+c)
    int d4    = (idx & 15) << 2;                         // 0,4,...,60
    int b     = bc >> 8;
    int c     = bc & 255;

    v4f z = *(const v4f*)(zbuf + (size_t)b * (CENC * DBINS) + c * DBINS + d4);

    const float* epos = enc + (size_t)bc * 256;
    int hw0 = chunk * 64;
    for (int hw = hw0; hw < hw0 + 64; ++hw) {
        float e = epos[hw];
        v4f o;
        o.x = e * z.x; o.y = e * z.y; o.z = e * z.z; o.w = e * z.w;
        __builtin_nontemporal_store(
            o, (v4f*)(out + ((size_t)bc * 256 + hw) * DBINS + d4));
    }
}

// ---------------------------------------------------------------------------
extern "C" void kernel_launch(void* const* d_in, const int* in_sizes, int n_in,
                              void* d_out, int out_size, void* d_ws, size_t ws_size,
                              hipStream_t stream) {
    const float* img   = (const float*)d_in[0];
    const float* enc   = (const float*)d_in[1];
    const float* wx    = (const float*)d_in[2];
    const float* bx    = (const float*)d_in[3];
    const float* wy    = (const float*)d_in[4];
    const float* by    = (const float*)d_in[5];
    const float* wz1   = (const float*)d_in[6];
    const float* bz1   = (const float*)d_in[7];
    const float* gamma = (const float*)d_in[8];
    const float* beta  = (const float*)d_in[9];
    const float* rmean = (const float*)d_in[10];
    const float* rvar  = (const float*)d_in[11];
    const float* wz2   = (const float*)d_in[12];
    const float* bz2   = (const float*)d_in[13];

    float* ws   = (float*)d_ws;
    float* fx   = ws;                  // 16*256*16  = 65536
    float* fy   = ws + 65536;          // 65536
    float* fz   = ws + 131072;         // 16*512     = 8192
    float* zbuf = ws + 139264;         // 16*16384   = 262144

    float* out_coord = (float*)d_out;              // [16,21,3] = 1008
    float* out_xyz   = out_coord + NB * JNT * 3;   // [16,256,16,16,64]

    k_enc_means<<<NB * CENC, 32, 0, stream>>>(enc, fx, fy);
    k_fz<<<(NB * CIMG + 255) / 256, 256, 0, stream>>>(img, fz);
    k_zgemm<<<128, 256, 0, stream>>>(wz1, fz, bz1, gamma, beta, rmean, rvar, zbuf);
    k_coords<<<NB * JNT, 128, 0, stream>>>(fx, fy, zbuf, wx, bx, wy, by, wz2, bz2,
                                           out_coord);
    k_bcast<<<1024, 256, 0, stream>>>(enc, zbuf, out_xyz);
}